// GAT_8684423873163
// MI455X (gfx1250) — compile-verified
//
#include <hip/hip_runtime.h>
#include <math.h>

#define NN   100000
#define EE   1000000
#define E2   (EE + NN)
#define GG   64
#define HID  64

typedef __attribute__((ext_vector_type(2))) float v2f;
typedef __attribute__((ext_vector_type(8))) float v8f;

// ---- ordered-int encoding so atomicMax(uint) == float max ----
__device__ __forceinline__ unsigned ordf(float f) {
    unsigned u = __float_as_uint(f);
    return (u & 0x80000000u) ? ~u : (u | 0x80000000u);
}
__device__ __forceinline__ float unordf(unsigned u) {
    return __uint_as_float((u & 0x80000000u) ? (u & 0x7fffffffu) : ~u);
}

__global__ void fill_u32(unsigned* __restrict__ p, unsigned v, int n) {
    int i = blockIdx.x * blockDim.x + threadIdx.x;
    if (i < n) p[i] = v;
}

// ---------------- WMMA fp32 GEMM:  H[N,64] = X[N,K] @ W[K,64] ----------------
// wave -> 16x16 C tile; block = 8 waves = 32 rows x 64 cols; W staged in LDS.
template <int K>
__global__ void gemm_wmma(const float* __restrict__ X, const float* __restrict__ W,
                          float* __restrict__ H) {
    __shared__ float sB[K * 64];
    for (int i = threadIdx.x; i < K * 64; i += 256) sB[i] = W[i];
    __syncthreads();

    const int wave  = threadIdx.x >> 5;
    const int lane  = threadIdx.x & 31;
    const int mtile = wave >> 2;          // 0..1
    const int ntile = wave & 3;           // 0..3
    const int g     = lane >> 4;          // K-half select
    const int l16   = lane & 15;
    const int m     = blockIdx.x * 32 + mtile * 16 + l16;  // A row (M = lane%16)
    const int n     = ntile * 16 + l16;                    // B col (N = lane%16)

    v8f acc = {};
    const float* xrow = X + (long)m * K;
    for (int k0 = 0; k0 < K; k0 += 4) {
        const int kk = k0 + 2 * g;                 // lanes 0-15: K={k0,k0+1}; 16-31: {k0+2,k0+3}
        v2f a = *(const v2f*)(xrow + kk);
        v2f b;
        b.x = sB[kk * 64 + n];
        b.y = sB[(kk + 1) * 64 + n];
        acc = __builtin_amdgcn_wmma_f32_16x16x4_f32(false, a, false, b,
                                                    (short)0, acc, false, false);
    }
    // C layout: VGPR r -> M = r + 8*g, N = lane%16
    float* out = H + (long)(blockIdx.x * 32 + mtile * 16 + g * 8) * 64 + ntile * 16 + l16;
#pragma unroll
    for (int r = 0; r < 8; ++r) out[(long)r * 64] = acc[r];
}

// ---- per-node attention scalars: as[n]=h[n]·a_src, ad[n]=h[n]·a_dst (wave/node)
__global__ void alpha_kernel(const float* __restrict__ H, const float* __restrict__ av_s,
                             const float* __restrict__ av_d, float* __restrict__ as_o,
                             float* __restrict__ ad_o) {
    const int wave = threadIdx.x >> 5, lane = threadIdx.x & 31;
    const int node = blockIdx.x * 8 + wave;
    const float* hr = H + (long)node * 64;
    float h0 = hr[lane], h1 = hr[lane + 32];
    float ps = h0 * av_s[lane] + h1 * av_s[lane + 32];
    float pd = h0 * av_d[lane] + h1 * av_d[lane + 32];
    for (int off = 16; off > 0; off >>= 1) {
        ps += __shfl_xor(ps, off, 32);
        pd += __shfl_xor(pd, off, 32);
    }
    if (lane == 0) { as_o[node] = ps; ad_o[node] = pd; }
}

__device__ __forceinline__ void edge_sd(const int* __restrict__ adj, int e, int& s, int& d) {
    if (e < EE) { s = adj[e]; d = adj[EE + e]; }
    else        { s = e - EE; d = s; }          // appended self loops
}

__global__ void edge_score(const int* __restrict__ adj, const float* __restrict__ as_,
                           const float* __restrict__ ad_, float* __restrict__ esc,
                           unsigned* __restrict__ mx) {
    int e = blockIdx.x * blockDim.x + threadIdx.x;
    if (e >= E2) return;
    int s, d; edge_sd(adj, e, s, d);
    float v = as_[s] + ad_[d];
    v = (v > 0.f) ? v : 0.2f * v;               // leaky relu
    esc[e] = v;
    atomicMax(mx + d, ordf(v));
}

__global__ void edge_exp(const int* __restrict__ adj, float* __restrict__ esc,
                         const unsigned* __restrict__ mx, float* __restrict__ den) {
    int e = blockIdx.x * blockDim.x + threadIdx.x;
    if (e >= E2) return;
    int s, d; edge_sd(adj, e, s, d);
    float num = expf(esc[e] - unordf(mx[d]));
    esc[e] = num;                               // in place: score -> numerator
    atomicAdd(den + d, num);
}

__global__ void edge_coef(const int* __restrict__ adj, float* __restrict__ esc,
                          const float* __restrict__ den) {
    int e = blockIdx.x * blockDim.x + threadIdx.x;
    if (e >= E2) return;
    int s, d; edge_sd(adj, e, s, d);
    esc[e] = esc[e] / (den[d] + 1e-16f);        // in place: numerator -> coef
}

// scatter-add: 16 threads/edge, float4 each; gathers hit the 192MB L2 (h = 25.6MB)
__global__ void agg_kernel(const int* __restrict__ adj, const float* __restrict__ coef,
                           const float* __restrict__ H, float* __restrict__ A) {
    int idx = blockIdx.x * blockDim.x + threadIdx.x;  // E2*16 threads
    int e = idx >> 4, q = (idx & 15) << 2;
    int s, d; edge_sd(adj, e, s, d);
    float c = coef[e];
    const float4 hv = *(const float4*)(H + (long)s * 64 + q);
    float* out = A + (long)d * 64 + q;
    atomicAdd(out + 0, c * hv.x);
    atomicAdd(out + 1, c * hv.y);
    atomicAdd(out + 2, c * hv.z);
    atomicAdd(out + 3, c * hv.w);
}

__global__ void bias_relu(const float* __restrict__ A, const float* __restrict__ b,
                          float* __restrict__ X) {
    int i = blockIdx.x * 256 + threadIdx.x;     // N*64
    float v = A[i] + b[i & 63];
    X[i] = (v > 0.f) ? v : 0.f;
}

__global__ void pool_kernel(const float* __restrict__ X, const int* __restrict__ batch,
                            unsigned* __restrict__ pooled) {
    int i = blockIdx.x * 256 + threadIdx.x;     // N*64
    int n = i >> 6, f = i & 63;
    atomicMax(pooled + batch[n] * 64 + f, ordf(X[i]));
}

__global__ void root_kernel(const int* __restrict__ batch, int* __restrict__ root) {
    int n = blockIdx.x * blockDim.x + threadIdx.x;
    if (n < NN) atomicMin(root + batch[n], n);
}

__global__ void lin0_kernel(const unsigned* __restrict__ pooled, const float* __restrict__ W,
                            const float* __restrict__ b, float* __restrict__ hg) {
    int i = blockIdx.x * blockDim.x + threadIdx.x;  // G*64
    int g = i >> 6, j = i & 63;
    const unsigned* p = pooled + g * 64;
    float acc = b[j];
    for (int k = 0; k < 64; ++k) acc += unordf(p[k]) * W[k * 64 + j];
    hg[i] = (acc > 0.f) ? acc : 0.f;
}

__global__ void news_kernel(const float* __restrict__ x, const int* __restrict__ root,
                            const float* __restrict__ W, const float* __restrict__ b,
                            float* __restrict__ news) {
    int i = blockIdx.x * blockDim.x + threadIdx.x;  // G*64
    int g = i >> 6, j = i & 63;
    int r = root[g];
    if ((unsigned)r >= NN) r = 0;
    const float* xr = x + (long)r * 128;
    float acc = b[j];
    for (int k = 0; k < 128; ++k) acc += xr[k] * W[k * 64 + j];
    news[i] = (acc > 0.f) ? acc : 0.f;
}

__global__ void final_kernel(const float* __restrict__ hg, const float* __restrict__ news,
                             const float* __restrict__ W, const float* __restrict__ b,
                             float* __restrict__ out) {
    int g = threadIdx.x;                        // G threads
    float acc = b[0];
    for (int j = 0; j < 64; ++j) acc += hg[g * 64 + j] * W[j];
    for (int j = 0; j < 64; ++j) acc += news[g * 64 + j] * W[64 + j];
    out[g] = 1.f / (1.f + expf(-acc));
}

extern "C" void kernel_launch(void* const* d_in, const int* in_sizes, int n_in,
                              void* d_out, int out_size, void* d_ws, size_t ws_size,
                              hipStream_t stream) {
    (void)in_sizes; (void)n_in; (void)out_size; (void)ws_size;
    const float* x      = (const float*)d_in[0];
    const int*   adj    = (const int*)d_in[1];
    const int*   batch  = (const int*)d_in[2];
    const float* W[3]   = {(const float*)d_in[3],  (const float*)d_in[7],  (const float*)d_in[11]};
    const float* avs[3] = {(const float*)d_in[4],  (const float*)d_in[8],  (const float*)d_in[12]};
    const float* avd[3] = {(const float*)d_in[5],  (const float*)d_in[9],  (const float*)d_in[13]};
    const float* bv[3]  = {(const float*)d_in[6],  (const float*)d_in[10], (const float*)d_in[14]};
    const float* linnews_W = (const float*)d_in[15];
    const float* linnews_b = (const float*)d_in[16];
    const float* lin0_W    = (const float*)d_in[17];
    const float* lin0_b    = (const float*)d_in[18];
    const float* lin1_W    = (const float*)d_in[19];
    const float* lin1_b    = (const float*)d_in[20];

    // workspace layout (floats): ~83 MB total
    float*    ws     = (float*)d_ws;
    float*    bufA   = ws;                      // N*64  layer input (relu'd)
    float*    bufH   = bufA   + (long)NN * 64;  // N*64  gemm output h
    float*    bufAgg = bufH   + (long)NN * 64;  // N*64  aggregation
    float*    as_    = bufAgg + (long)NN * 64;  // N
    float*    ad_    = as_    + NN;             // N
    float*    den    = ad_    + NN;             // N
    unsigned* mx     = (unsigned*)(den + NN);   // N
    float*    esc    = (float*)(mx + NN);       // E2
    unsigned* pooled = (unsigned*)(esc + E2);   // G*64
    float*    hg     = (float*)(pooled + GG * 64);
    float*    news   = hg + GG * 64;
    int*      root   = (int*)(news + GG * 64);

    const int eg = (E2 + 255) / 256;
    for (int l = 0; l < 3; ++l) {
        if (l == 0) gemm_wmma<128><<<NN / 32, 256, 0, stream>>>(x,    W[0], bufH);
        else        gemm_wmma<64> <<<NN / 32, 256, 0, stream>>>(bufA, W[l], bufH);
        alpha_kernel<<<NN / 8, 256, 0, stream>>>(bufH, avs[l], avd[l], as_, ad_);
        fill_u32<<<(NN + 255) / 256, 256, 0, stream>>>(mx, 0u, NN);
        fill_u32<<<(NN + 255) / 256, 256, 0, stream>>>((unsigned*)den, 0u, NN);
        fill_u32<<<(NN * 64) / 256, 256, 0, stream>>>((unsigned*)bufAgg, 0u, NN * 64);
        edge_score<<<eg, 256, 0, stream>>>(adj, as_, ad_, esc, mx);
        edge_exp  <<<eg, 256, 0, stream>>>(adj, esc, mx, den);
        edge_coef <<<eg, 256, 0, stream>>>(adj, esc, den);
        agg_kernel<<<(E2 * 16) / 256, 256, 0, stream>>>(adj, esc, bufH, bufAgg);
        bias_relu <<<(NN * 64) / 256, 256, 0, stream>>>(bufAgg, bv[l], bufA);
    }
    fill_u32<<<(GG * 64 + 255) / 256, 256, 0, stream>>>(pooled, 0u, GG * 64);
    fill_u32<<<1, 64, 0, stream>>>((unsigned*)root, 0x7fffffffu, GG);
    pool_kernel<<<(NN * 64) / 256, 256, 0, stream>>>(bufA, batch, pooled);
    root_kernel<<<(NN + 255) / 256, 256, 0, stream>>>(batch, root);
    lin0_kernel<<<GG, 64, 0, stream>>>(pooled, lin0_W, lin0_b, hg);
    news_kernel<<<GG, 64, 0, stream>>>(x, root, linnews_W, linnews_b, news);
    final_kernel<<<1, GG, 0, stream>>>(hg, news, lin1_W, lin1_b, (float*)d_out);
}